// IBDCRF_13692355740456
// MI455X (gfx1250) — compile-verified
//
#include <hip/hip_runtime.h>
#include <hip/hip_bf16.h>
#include <math.h>

#define B_ 32
#define T_ 8192
#define D_ 128
#define CHUNK 128
#define NC (T_ / CHUNK)      // 64 chunks per sequence
#define NTASK (B_ * NC)      // 2048 chunk tasks
#define NEGF (-1.0e30f)
#define FMIN (-3.4028234663852886e38f)   // float32 finfo.min (reference sentinel)

typedef __attribute__((ext_vector_type(2))) float v2f;
typedef __attribute__((ext_vector_type(8))) float v8f;
typedef __attribute__((ext_vector_type(4))) int   v4i;

#define AS1_ __attribute__((address_space(1)))
#define AS3_ __attribute__((address_space(3)))

#if defined(__has_builtin)
#if __has_builtin(__builtin_amdgcn_global_load_async_to_lds_b128) && \
    __has_builtin(__builtin_amdgcn_s_wait_asynccnt)
#define USE_ASYNC_LDS 1
#endif
#endif

// ---------- helpers ----------
__device__ __forceinline__ float lse2f(float a, float b) {
  float m = fmaxf(a, b);
  return m + log1pf(__expf(-fabsf(a - b)));   // arg <= 0, native exp is ~1ulp here
}

// log transition matrix for edge with distance d (row = from, col = to)
__device__ __forceinline__ void make_trans(float d, float tr[4]) {
  d = fmaxf(d, 1e-6f);
  float e0 = -0.02f * d;   // RATE_01
  float e1 = -0.5f  * d;   // RATE_10
  tr[0] = e0;                  // log P(0->0)
  tr[1] = logf(-expm1f(e0));   // log P(0->1)
  tr[2] = logf(-expm1f(e1));   // log P(1->0)
  tr[3] = e1;                  // log P(1->1)
}

// ---------- K1: emission potentials via V_WMMA_F32_16X16X4_F32 ----------
// pot[b,t,s] = feats[b,t,:] @ W[:,s] + b[s].  4 waves/block, 1 16-row tile/wave.
__global__ __launch_bounds__(128) void emit_kernel(const float* __restrict__ feats,
                                                   const float* __restrict__ W,
                                                   const float* __restrict__ bvec,
                                                   float* __restrict__ pot) {
  __shared__ float tileA[4][16][132];   // pad 132: stride%64==4 -> conflict-free, 16B aligned
  __shared__ float Bl[64][16][2];       // pre-expanded B tiles: Bl[kpair][n][j] = W[(2*kp+j)*2+n] (n<2) else 0
  __shared__ float bl[2];
  const int tid = threadIdx.x;
  // build B tiles once per block: unconditional reads in the WMMA loop afterwards
  for (int i = tid; i < 64 * 16; i += 128) {
    const int kp = i >> 4, n = i & 15;
    Bl[kp][n][0] = (n < 2) ? W[(2 * kp) * 2 + n]     : 0.0f;
    Bl[kp][n][1] = (n < 2) ? W[(2 * kp + 1) * 2 + n] : 0.0f;
  }
  if (tid < 2) bl[tid] = bvec[tid];
  const int w    = tid >> 5;
  const int lane = tid & 31;
  const long tile    = (long)blockIdx.x * 4 + w;
  const long rowBase = tile * 16;
  // coalesced stage of the 16x128 f32 A tile
#ifdef USE_ASYNC_LDS
  for (int r = 0; r < 16; ++r) {
    __builtin_amdgcn_global_load_async_to_lds_b128(
        (AS1_ v4i*)(feats + (rowBase + r) * D_ + lane * 4),
        (AS3_ v4i*)&tileA[w][r][lane * 4],
        0, 0);
  }
  __builtin_amdgcn_s_wait_asynccnt(0);
#else
  for (int r = 0; r < 16; ++r) {
    const float4 v = *(const float4*)(feats + (rowBase + r) * D_ + lane * 4);
    *(float4*)&tileA[w][r][lane * 4] = v;
  }
#endif
  __syncthreads();
  const int m   = lane & 15;   // A row
  const int n   = lane & 15;   // B/D column
  const int grp = lane >> 4;   // k-pair selector (0 -> k{0,1}, 1 -> k{2,3})
  v8f c = {};
  #pragma unroll
  for (int k4 = 0; k4 < D_; k4 += 4) {
    const int ka = k4 + grp * 2;
    const int kp = (k4 >> 1) + grp;
    v2f a;
    a.x = tileA[w][m][ka];
    a.y = tileA[w][m][ka + 1];
    v2f bm = *(const v2f*)&Bl[kp][n][0];
    // D = A(16x4, f32) * B(4x16, f32) + C  -- exact f32 accumulate
    c = __builtin_amdgcn_wmma_f32_16x16x4_f32(false, a, false, bm, (short)0, c, false, false);
  }
  if (n < 2) {   // only columns 0,1 carry real data
    const long g0 = rowBase + grp * 8;   // VGPR v holds M = v (+8 for upper lane group)
    const float bb = bl[n];
    #pragma unroll
    for (int v = 0; v < 8; ++v) pot[(g0 + v) * 2 + n] = c[v] + bb;
  }
}

// ---------- K2: lengths[b] = sum(mask[b,:]) ----------
__global__ __launch_bounds__(256) void lengths_kernel(const unsigned char* __restrict__ mask,
                                                      int* __restrict__ lens) {
  __shared__ int sdata[256];
  const int b = blockIdx.x;
  int s = 0;
  for (int i = threadIdx.x; i < T_; i += 256) s += mask[(long)b * T_ + i] ? 1 : 0;
  sdata[threadIdx.x] = s;
  __syncthreads();
  for (int o = 128; o > 0; o >>= 1) {
    if (threadIdx.x < o) sdata[threadIdx.x] += sdata[threadIdx.x + o];
    __syncthreads();
  }
  if (threadIdx.x == 0) lens[b] = sdata[0];
}

// ---------- K3: forward pass 1 — per-chunk 2x2 log-matrix product ----------
// step matrix M_t[i][j] = trans_t[i][j] + pot_t[j] (NEG matrix when masked)
__global__ __launch_bounds__(256) void fwd_pass1(const float* __restrict__ pot,
                                                 const float* __restrict__ dcm,
                                                 const unsigned char* __restrict__ mask,
                                                 float* __restrict__ P) {
  const int task = blockIdx.x * blockDim.x + threadIdx.x;
  if (task >= NTASK) return;
  const int b = task / NC, c = task % NC;
  const long base = (long)b * T_;
  float p0 = 0.0f, p1 = NEGF, p2 = NEGF, p3 = 0.0f;   // log-identity
  int t1 = (c + 1) * CHUNK; if (t1 > T_ - 1) t1 = T_ - 1;
  for (int t = c * CHUNK + 1; t <= t1; ++t) {
    float M0, M1, M2, M3;
    if (mask[base + t]) {
      float tr[4]; make_trans(dcm[base + t], tr);
      const float q0 = pot[(base + t) * 2], q1 = pot[(base + t) * 2 + 1];
      M0 = tr[0] + q0; M1 = tr[1] + q1; M2 = tr[2] + q0; M3 = tr[3] + q1;
    } else { M0 = M1 = M2 = M3 = NEGF; }
    const float n0 = lse2f(p0 + M0, p1 + M2);
    const float n1 = lse2f(p0 + M1, p1 + M3);
    const float n2 = lse2f(p2 + M0, p3 + M2);
    const float n3 = lse2f(p2 + M1, p3 + M3);
    p0 = n0; p1 = n1; p2 = n2; p3 = n3;
  }
  P[task * 4 + 0] = p0; P[task * 4 + 1] = p1;
  P[task * 4 + 2] = p2; P[task * 4 + 3] = p3;
}

// ---------- K4: forward pass 2 — scan chunk boundaries (alpha at t=c*CHUNK) ----------
__global__ void fwd_pass2(const float* __restrict__ pot,
                          const unsigned char* __restrict__ mask,
                          const float* __restrict__ P,
                          float* __restrict__ Vb) {
  const int b = threadIdx.x;
  if (b >= B_) return;
  const long base = (long)b * T_;
  float v0 = mask[base] ? pot[base * 2]     : NEGF;
  float v1 = mask[base] ? pot[base * 2 + 1] : NEGF;
  for (int c = 0; c < NC; ++c) {
    Vb[(b * NC + c) * 2]     = v0;
    Vb[(b * NC + c) * 2 + 1] = v1;
    const float* p = P + (long)(b * NC + c) * 4;
    const float n0 = lse2f(v0 + p[0], v1 + p[2]);
    const float n1 = lse2f(v0 + p[1], v1 + p[3]);
    v0 = n0; v1 = n1;
  }
}

// ---------- K5: forward pass 3 — replay chunks, write alpha into out_lp ----------
__global__ __launch_bounds__(256) void fwd_pass3(const float* __restrict__ pot,
                                                 const float* __restrict__ dcm,
                                                 const unsigned char* __restrict__ mask,
                                                 const float* __restrict__ Vb,
                                                 float* __restrict__ out_lp) {
  const int task = blockIdx.x * blockDim.x + threadIdx.x;
  if (task >= NTASK) return;
  const int b = task / NC, c = task % NC;
  const long base = (long)b * T_;
  float v0 = Vb[(b * NC + c) * 2];
  float v1 = Vb[(b * NC + c) * 2 + 1];
  const int t0 = c * CHUNK;
  out_lp[(base + t0) * 2]     = v0;
  out_lp[(base + t0) * 2 + 1] = v1;
  for (int t = t0 + 1; t <= t0 + CHUNK - 1; ++t) {
    float n0, n1;
    if (mask[base + t]) {
      float tr[4]; make_trans(dcm[base + t], tr);
      n0 = lse2f(v0 + tr[0], v1 + tr[2]) + pot[(base + t) * 2];
      n1 = lse2f(v0 + tr[1], v1 + tr[3]) + pot[(base + t) * 2 + 1];
    } else { n0 = n1 = NEGF; }
    v0 = n0; v1 = n1;
    out_lp[(base + t) * 2]     = v0;
    out_lp[(base + t) * 2 + 1] = v1;
  }
}

// ---------- K6: backward pass 1 — per-chunk affine (A, c): beta_lo = A (+) beta_hi (lse+) c ----------
__global__ __launch_bounds__(256) void bwd_pass1(const float* __restrict__ pot,
                                                 const float* __restrict__ dcm,
                                                 const unsigned char* __restrict__ mask,
                                                 const int* __restrict__ lens,
                                                 float* __restrict__ Abw,
                                                 float* __restrict__ cbw) {
  (void)mask;
  const int task = blockIdx.x * blockDim.x + threadIdx.x;
  if (task >= NTASK) return;
  const int b = task / NC, c = task % NC;
  const int len = lens[b];
  const long base = (long)b * T_;
  float A0 = 0.0f, A1 = NEGF, A2 = NEGF, A3 = 0.0f;   // log-identity
  float c0 = NEGF, c1 = NEGF;
  int top = (c + 1) * CHUNK; if (top > T_ - 1) top = T_ - 1;
  for (int t = top - 1; t >= c * CHUNK; --t) {
    if (t >= len)          { A0 = A1 = A2 = A3 = NEGF; c0 = NEGF; c1 = NEGF; }
    else if (t == len - 1) { A0 = A1 = A2 = A3 = NEGF; c0 = 0.0f; c1 = 0.0f; }
    else {
      float tr[4]; make_trans(dcm[base + t + 1], tr);
      const float q0 = pot[(base + t + 1) * 2], q1 = pot[(base + t + 1) * 2 + 1];
      const float M0 = tr[0] + q0, M1 = tr[1] + q1, M2 = tr[2] + q0, M3 = tr[3] + q1;
      const float n0 = lse2f(M0 + A0, M1 + A2);
      const float n1 = lse2f(M0 + A1, M1 + A3);
      const float n2 = lse2f(M2 + A0, M3 + A2);
      const float n3 = lse2f(M2 + A1, M3 + A3);
      const float d0 = lse2f(M0 + c0, M1 + c1);
      const float d1 = lse2f(M2 + c0, M3 + c1);
      A0 = n0; A1 = n1; A2 = n2; A3 = n3; c0 = d0; c1 = d1;
    }
  }
  Abw[task * 4 + 0] = A0; Abw[task * 4 + 1] = A1;
  Abw[task * 4 + 2] = A2; Abw[task * 4 + 3] = A3;
  cbw[task * 2 + 0] = c0; cbw[task * 2 + 1] = c1;
}

// ---------- K7: backward pass 2 — ll[b] and beta chunk-top boundaries ----------
__global__ void bwd_pass2(const float* __restrict__ alpha_lp,
                          const int* __restrict__ lens,
                          const float* __restrict__ Abw,
                          const float* __restrict__ cbw,
                          float* __restrict__ Rtop,
                          float* __restrict__ llw,
                          float* __restrict__ out_ll) {
  const int b = threadIdx.x;
  if (b >= B_) return;
  const int len = lens[b];
  const long base = (long)b * T_;
  float ll;
  if (len > 0) ll = lse2f(alpha_lp[(base + len - 1) * 2], alpha_lp[(base + len - 1) * 2 + 1]);
  else         ll = FMIN;
  llw[b] = ll; out_ll[b] = ll;
  float v0, v1;   // beta at t = T-1 (= bT)
  if (len == T_) { v0 = 0.0f; v1 = 0.0f; } else { v0 = NEGF; v1 = NEGF; }
  for (int c = NC - 1; c >= 0; --c) {
    Rtop[(b * NC + c) * 2]     = v0;   // beta at the top boundary feeding chunk c
    Rtop[(b * NC + c) * 2 + 1] = v1;
    const float* A  = Abw + (long)(b * NC + c) * 4;
    const float* cc = cbw + (long)(b * NC + c) * 2;
    const float n0 = lse2f(lse2f(A[0] + v0, A[1] + v1), cc[0]);
    const float n1 = lse2f(lse2f(A[2] + v0, A[3] + v1), cc[1]);
    v0 = n0; v1 = n1;
  }
}

__device__ __forceinline__ void finalize_t(float* __restrict__ out_lp, float* __restrict__ out_post,
                                           long base, int t, float v0, float v1, float ll, bool mk) {
  const long i = (base + t) * 2;
  float lp0, lp1;
  if (mk) { lp0 = out_lp[i] + v0 - ll; lp1 = out_lp[i + 1] + v1 - ll; }
  else    { lp0 = FMIN;                lp1 = FMIN; }
  out_lp[i] = lp0;  out_lp[i + 1] = lp1;
  out_post[i] = expf(lp0);  out_post[i + 1] = expf(lp1);
}

// ---------- K8: backward pass 3 — replay beta, fuse log_post + posterior ----------
__global__ __launch_bounds__(256) void bwd_pass3(const float* __restrict__ pot,
                                                 const float* __restrict__ dcm,
                                                 const unsigned char* __restrict__ mask,
                                                 const int* __restrict__ lens,
                                                 const float* __restrict__ Rtop,
                                                 const float* __restrict__ llw,
                                                 float* __restrict__ out_lp,
                                                 float* __restrict__ out_post) {
  const int task = blockIdx.x * blockDim.x + threadIdx.x;
  if (task >= NTASK) return;
  const int b = task / NC, c = task % NC;
  const int len = lens[b];
  const float ll = llw[b];
  const long base = (long)b * T_;
  float v0 = Rtop[(b * NC + c) * 2];
  float v1 = Rtop[(b * NC + c) * 2 + 1];
  if (c == NC - 1) finalize_t(out_lp, out_post, base, T_ - 1, v0, v1, ll, mask[base + T_ - 1] != 0);
  int top = (c + 1) * CHUNK; if (top > T_ - 1) top = T_ - 1;
  for (int t = top - 1; t >= c * CHUNK; --t) {
    if (t >= len)          { v0 = NEGF; v1 = NEGF; }
    else if (t == len - 1) { v0 = 0.0f; v1 = 0.0f; }
    else {
      float tr[4]; make_trans(dcm[base + t + 1], tr);
      const float q0 = pot[(base + t + 1) * 2], q1 = pot[(base + t + 1) * 2 + 1];
      const float M0 = tr[0] + q0, M1 = tr[1] + q1, M2 = tr[2] + q0, M3 = tr[3] + q1;
      const float n0 = lse2f(M0 + v0, M1 + v1);
      const float n1 = lse2f(M2 + v0, M3 + v1);
      v0 = n0; v1 = n1;
    }
    finalize_t(out_lp, out_post, base, t, v0, v1, ll, mask[base + t] != 0);
  }
}

// ---------- launcher ----------
extern "C" void kernel_launch(void* const* d_in, const int* in_sizes, int n_in,
                              void* d_out, int out_size, void* d_ws, size_t ws_size,
                              hipStream_t stream) {
  (void)in_sizes; (void)n_in; (void)out_size; (void)ws_size;
  const float* feats        = (const float*)d_in[0];
  const float* dcm          = (const float*)d_in[1];
  const unsigned char* mask = (const unsigned char*)d_in[2];
  const float* W            = (const float*)d_in[3];
  const float* bvec         = (const float*)d_in[4];

  float* out_lp   = (float*)d_out;                 // [B,T,2] log_post (holds alpha transiently)
  float* out_post = out_lp + (long)B_ * T_ * 2;    // [B,T,2] posterior
  float* out_ll   = out_post + (long)B_ * T_ * 2;  // [B]

  float* ws   = (float*)d_ws;
  float* pot  = ws;                         // B*T*2
  float* P    = pot + (long)B_ * T_ * 2;    // NTASK*4
  float* Vb   = P + (long)NTASK * 4;        // NTASK*2
  float* Abw  = Vb + (long)NTASK * 2;       // NTASK*4
  float* cbw  = Abw + (long)NTASK * 4;      // NTASK*2
  float* Rtop = cbw + (long)NTASK * 2;      // NTASK*2
  float* llw  = Rtop + (long)NTASK * 2;     // B
  int*   lens = (int*)(llw + B_);           // B

  emit_kernel<<<(B_ * T_ / 16) / 4, 128, 0, stream>>>(feats, W, bvec, pot);
  lengths_kernel<<<B_, 256, 0, stream>>>(mask, lens);
  fwd_pass1<<<NTASK / 256, 256, 0, stream>>>(pot, dcm, mask, P);
  fwd_pass2<<<1, B_, 0, stream>>>(pot, mask, P, Vb);
  fwd_pass3<<<NTASK / 256, 256, 0, stream>>>(pot, dcm, mask, Vb, out_lp);
  bwd_pass1<<<NTASK / 256, 256, 0, stream>>>(pot, dcm, mask, lens, Abw, cbw);
  bwd_pass2<<<1, B_, 0, stream>>>(out_lp, lens, Abw, cbw, Rtop, llw, out_ll);
  bwd_pass3<<<NTASK / 256, 256, 0, stream>>>(pot, dcm, mask, lens, Rtop, llw, out_lp, out_post);
}